// PointerNetwork_53240414601480
// MI455X (gfx1250) — compile-verified
//
#include <hip/hip_runtime.h>
#include <hip/hip_bf16.h>
#include <stdint.h>

// ---------------------------------------------------------------------------
// PointerNetwork additive attention for MI455X (gfx1250, wave32)
//   enc_proj = enc @ W1^T   (WMMA f32 16x16x4)
//   dec_proj = dec @ W2^T   (WMMA f32 16x16x4)
//   scores[b,d,e] = sum_c v[c] * tanh(dec_proj[b,d,c] + enc_proj[b,e,c])
//   out = log_softmax(mask ? scores : -1e9, axis=e)
//
// Score kernel: 4 decoder rows per block (4x eproj L2-traffic reuse),
// eproj staged into LDS with async global->LDS DMA (ASYNCcnt), double-buffered.
// ---------------------------------------------------------------------------

typedef float v2f __attribute__((ext_vector_type(2)));
typedef float v8f __attribute__((ext_vector_type(8)));

#define NEG_INF_F (-1e9f)

// --------------------------------------------------------------------------
// Kernel 1: Y[b,m,n] = sum_k X[b,m,k] * W[n,k]   (W shared across batch)
// One wave computes one 16x16 tile via V_WMMA_F32_16X16X4_F32, K steps of 4.
// A (16x4): lanes 0-15 hold M=lane with K{0,1}; lanes 16-31 K{2,3}.
// B (4x16): lanes 0-15 hold N=lane with K{0,1}; lanes 16-31 K{2,3}.
// C/D (16x16 f32): VGPR j -> M = j + 8*(lane>=16), N = lane&15.
// --------------------------------------------------------------------------
__global__ __launch_bounds__(32)
void pn_proj_gemm_wmma(const float* __restrict__ X,  // [B, M, K]
                       const float* __restrict__ W,  // [N, K]
                       float* __restrict__ Y,        // [B, M, N]
                       int M, int K, int N)
{
    const int tileM = blockIdx.x;
    const int tileN = blockIdx.y;
    const int b     = blockIdx.z;
    const int lane  = threadIdx.x;       // 0..31, EXEC all-ones for WMMA
    const int half  = lane >> 4;
    const int l     = lane & 15;

    const int row0 = tileM * 16;
    const int col0 = tileN * 16;

    const float* Xb = X + (size_t)b * M * K;
    float*       Yb = Y + (size_t)b * M * N;

    const float* arow = Xb + (size_t)(row0 + l) * K + half * 2;
    const float* brow = W  + (size_t)(col0 + l) * K + half * 2;

    v8f acc = {};
    for (int k = 0; k < K; k += 4) {
        v2f a  = *(const v2f*)(arow + k);
        v2f bm = *(const v2f*)(brow + k);
        acc = __builtin_amdgcn_wmma_f32_16x16x4_f32(
            false, a, false, bm, (short)0, acc, false, false);
    }

#pragma unroll
    for (int j = 0; j < 8; ++j) {
        Yb[(size_t)(row0 + j + half * 8) * N + (col0 + l)] = acc[j];
    }
}

// --------------------------------------------------------------------------
// CDNA5 native transcendental tanh (TRANS pipe, co-executes with VALU).
// --------------------------------------------------------------------------
__device__ __forceinline__ float fast_tanh_f32(float x) {
    float r;
    asm("v_tanh_f32 %0, %1" : "=v"(r) : "v"(x));
    return r;
}

// --------------------------------------------------------------------------
// Kernel 2: block = (b, 4 decoder rows). Pipeline:
//   async-stage 8 e-rows of eproj into LDS (double buffer), compute
//   scores for 4 d-rows per staged element, then 4 masked log-softmaxes.
// --------------------------------------------------------------------------
#define DTILE   4
#define ECHUNK  8      // e-rows per staged LDS chunk (== waves per block)
#define CDIM    256
#define EDIM    512

__global__ __launch_bounds__(256)
void pn_score_logsoftmax(const float* __restrict__ dproj,  // [B,D,C]
                         const float* __restrict__ eproj,  // [B,E,C]
                         const uint8_t* __restrict__ mask, // [B,D,E] (bool)
                         const float* __restrict__ v,      // [C]
                         float* __restrict__ out,          // [B,D,E]
                         int B, int D)
{
    const int dblks = D / DTILE;
    const int b     = blockIdx.x / dblks;
    const int d0    = (blockIdx.x % dblks) * DTILE;
    const int t     = threadIdx.x;
    const int wave  = t >> 5;            // 0..7
    const int lane  = t & 31;

    __shared__ float dp_s[DTILE][CDIM];            // 4 KB
    __shared__ float v_s[CDIM];                    // 1 KB
    __shared__ float ep_buf[2][ECHUNK][CDIM];      // 16 KB, double buffered
    __shared__ float s_s[DTILE][EDIM];             // 8 KB
    __shared__ float red[256];                     // 1 KB

    // ---- load decoder rows + v into LDS ----
    for (int idx = t; idx < DTILE * CDIM; idx += blockDim.x) {
        const int dd = idx / CDIM, c = idx % CDIM;
        dp_s[dd][c] = dproj[((size_t)b * D + d0 + dd) * CDIM + c];
    }
    for (int c = t; c < CDIM; c += blockDim.x) v_s[c] = v[c];
    __syncthreads();

    // ---- hoist per-lane v and dp slices into registers ----
    float vreg[8], dpreg[DTILE][8];
#pragma unroll
    for (int j = 0; j < 8; ++j) {
        const int c = lane + 32 * j;
        vreg[j] = v_s[c];
#pragma unroll
        for (int dd = 0; dd < DTILE; ++dd) dpreg[dd][j] = dp_s[dd][c];
    }

    const float* ep_b = eproj + (size_t)b * EDIM * CDIM;

    // ---- stage chunk 0 into buffer 0 (async global->LDS DMA) ----
    {
#pragma unroll
        for (int u = t; u < ECHUNK * CDIM / 4; u += 256) {   // 16B units
            const int row = u >> 4;
            const int c4  = (u & 15) << 2;
            const float* g = ep_b + (size_t)row * CDIM + c4;
            const uint32_t lds = (uint32_t)(uintptr_t)&ep_buf[0][row][c4];
            asm volatile("global_load_async_to_lds_b128 %0, %1, off"
                         :: "v"(lds), "v"(g) : "memory");
        }
    }

    int cur = 0;
    const int nchunks = EDIM / ECHUNK;   // 64
    for (int chunk = 0; chunk < nchunks; ++chunk) {
        asm volatile("s_wait_asynccnt 0" ::: "memory");
        __syncthreads();   // chunk `cur` resident for all waves

        // issue next chunk into the other buffer while we compute
        if (chunk + 1 < nchunks) {
            const int e0n = (chunk + 1) * ECHUNK;
#pragma unroll
            for (int u = t; u < ECHUNK * CDIM / 4; u += 256) {
                const int row = u >> 4;
                const int c4  = (u & 15) << 2;
                const float* g = ep_b + (size_t)(e0n + row) * CDIM + c4;
                const uint32_t lds = (uint32_t)(uintptr_t)&ep_buf[cur ^ 1][row][c4];
                asm volatile("global_load_async_to_lds_b128 %0, %1, off"
                             :: "v"(lds), "v"(g) : "memory");
            }
        }

        // ---- compute: wave w owns e-row w of the chunk; 4 d-rows reuse ep ----
        const int e = chunk * ECHUNK + wave;
        const float* eprow = &ep_buf[cur][wave][0];
        float a0 = 0.f, a1 = 0.f, a2 = 0.f, a3 = 0.f;
#pragma unroll
        for (int j = 0; j < 8; ++j) {
            const float epv = eprow[lane + 32 * j];
            a0 = __builtin_fmaf(vreg[j], fast_tanh_f32(dpreg[0][j] + epv), a0);
            a1 = __builtin_fmaf(vreg[j], fast_tanh_f32(dpreg[1][j] + epv), a1);
            a2 = __builtin_fmaf(vreg[j], fast_tanh_f32(dpreg[2][j] + epv), a2);
            a3 = __builtin_fmaf(vreg[j], fast_tanh_f32(dpreg[3][j] + epv), a3);
        }
        // wave32 butterfly reductions
#pragma unroll
        for (int off = 16; off > 0; off >>= 1) {
            a0 += __shfl_xor(a0, off, 32);
            a1 += __shfl_xor(a1, off, 32);
            a2 += __shfl_xor(a2, off, 32);
            a3 += __shfl_xor(a3, off, 32);
        }
        if (lane == 0) {
            s_s[0][e] = a0; s_s[1][e] = a1; s_s[2][e] = a2; s_s[3][e] = a3;
        }
        __syncthreads();   // done reading ep_buf[cur] before it is refilled
        cur ^= 1;
    }

    // ---- masked log-softmax per decoder row ----
    for (int dd = 0; dd < DTILE; ++dd) {
        const int d = d0 + dd;
        const uint8_t* mrow = mask + ((size_t)b * D + d) * EDIM;

        float m = NEG_INF_F;
        for (int e = t; e < EDIM; e += blockDim.x) {
            const float val = mrow[e] ? s_s[dd][e] : NEG_INF_F;
            s_s[dd][e] = val;
            m = fmaxf(m, val);
        }
        red[t] = m;
        __syncthreads();
#pragma unroll
        for (int s = 128; s > 0; s >>= 1) {
            if (t < s) red[t] = fmaxf(red[t], red[t + s]);
            __syncthreads();
        }
        const float rowmax = red[0];
        __syncthreads();

        float sum = 0.f;
        for (int e = t; e < EDIM; e += blockDim.x)
            sum += __expf(s_s[dd][e] - rowmax);
        red[t] = sum;
        __syncthreads();
#pragma unroll
        for (int s = 128; s > 0; s >>= 1) {
            if (t < s) red[t] += red[t + s];
            __syncthreads();
        }
        const float logZ = __logf(red[0]);
        __syncthreads();

        float* orow = out + ((size_t)b * D + d) * EDIM;
        for (int e = t; e < EDIM; e += blockDim.x)
            orow[e] = s_s[dd][e] - rowmax - logZ;
        __syncthreads();
    }
}

// --------------------------------------------------------------------------
extern "C" void kernel_launch(void* const* d_in, const int* in_sizes, int n_in,
                              void* d_out, int out_size, void* d_ws, size_t ws_size,
                              hipStream_t stream)
{
    const float*   decoder_out = (const float*)d_in[0];   // [B,D,C]
    const float*   encoder_out = (const float*)d_in[1];   // [B,E,C]
    const uint8_t* mask        = (const uint8_t*)d_in[2]; // [B,D,E] bool
    const float*   W1          = (const float*)d_in[3];   // [C,C]
    const float*   W2          = (const float*)d_in[4];   // [C,C]
    const float*   v           = (const float*)d_in[5];   // [C]
    float*         out         = (float*)d_out;           // [B,D,E]

    const int B = 4, D = 256, E = 512, C = 256;

    float* eproj = (float*)d_ws;                      // [B,E,C] = 2 MB
    float* dproj = eproj + (size_t)B * E * C;         // [B,D,C] = 1 MB

    // encoder_proj = enc @ W1^T
    dim3 gE(E / 16, C / 16, B);
    pn_proj_gemm_wmma<<<gE, 32, 0, stream>>>(encoder_out, W1, eproj, E, C, C);
    // decoder_proj = dec @ W2^T
    dim3 gD(D / 16, C / 16, B);
    pn_proj_gemm_wmma<<<gD, 32, 0, stream>>>(decoder_out, W2, dproj, D, C, C);

    // additive-attention scores + masked log-softmax
    pn_score_logsoftmax<<<B * (D / DTILE), 256, 0, stream>>>(
        dproj, eproj, mask, v, out, B, D);
}